// Transformer_31988916420621
// MI455X (gfx1250) — compile-verified
//
#include <hip/hip_runtime.h>

// ---------------------------------------------------------------------------
// MI455X (gfx1250) transformer block, fp32 WMMA (V_WMMA_F32_16X16X4_F32).
// Compute-bound (~36 GFLOP vs ~130MB HBM traffic). GEMMs: 2x2 register
// blocking, LDS K-chunks staged by double-buffered async TDM DMAs
// (tensor_load_to_lds + s_wait_tensorcnt) with hardware row padding.
// Attention: flash-style online softmax, 64-wide shared K/V tiles,
// 2 barriers per 96 WMMAs.
// ---------------------------------------------------------------------------

#define CTC   1024
#define HEADS 16
#define NBAT  2
#define LSEQ  2048
#define DKH   32
#define DVH   64

typedef __attribute__((ext_vector_type(2))) float        v2f;
typedef __attribute__((ext_vector_type(8))) float        v8f;
typedef __attribute__((ext_vector_type(4))) unsigned int v4u;
typedef __attribute__((ext_vector_type(8))) int          v8i;
typedef __attribute__((ext_vector_type(4))) int          v4i;

__device__ __forceinline__ v8f wmma_f32(v2f a, v2f b, v8f c) {
    return __builtin_amdgcn_wmma_f32_16x16x4_f32(
        false, a, false, b, (short)0, c, false, false);
}

#if defined(__has_builtin)
#if __has_builtin(__builtin_amdgcn_tensor_load_to_lds) && \
    __has_builtin(__builtin_amdgcn_s_wait_tensorcnt)
#define USE_TDM 1
#endif
#endif

#ifdef USE_TDM
// 2D tile DMA: global (row stride strideX elements, fp32) -> LDS, with
// hardware padding of (padAmtEnc+1) DWORDs every 2^(padIntEnc+1) DWORDs.
// clang-23 signature: (u32x4 g0, i32x8 g1, i32x4 g2, i32x4 g3, i32x8 g4, i32 cpol)
__device__ __forceinline__ void tdm_load_2d_f32(
    unsigned ldsAddr, const float* gsrc,
    unsigned tileX, unsigned tileY, unsigned strideX,
    unsigned padIntEnc, unsigned padAmtEnc)
{
    const unsigned long long ga = (unsigned long long)(uintptr_t)gsrc;
    v4u g0;
    g0.x = 1u;                                   // count=1 (valid user D#)
    g0.y = ldsAddr;                              // lds_addr (bytes)
    g0.z = (unsigned)(ga & 0xffffffffu);         // global_addr[31:0]
    g0.w = (unsigned)((ga >> 32) & 0x01ffffffu)  // global_addr[56:32]
         | (2u << 30);                           // type = 2 ("image")
    v8i g1;
    g1[0] = (int)((2u << 16)                     // data_size = 4 bytes
                | (1u << 20)                     // pad_enable
                | (padIntEnc << 22)              // pad_interval
                | (padAmtEnc << 25));            // pad_amount
    g1[1] = (int)((tileX & 0xffffu) << 16);      // tensor_dim0[15:0] (=tileX)
    g1[2] = (int)((tileX >> 16) | ((tileY & 0xffffu) << 16)); // dim0 hi, dim1 lo
    g1[3] = (int)((tileY >> 16) | (tileX << 16));             // dim1 hi, tile_dim0
    g1[4] = (int)(tileY & 0xffffu);              // tile_dim1 (tile_dim2 = 0)
    g1[5] = (int)strideX;                        // tensor_dim0_stride[31:0]
    g1[6] = 0;                                   // stride hi bits
    g1[7] = 0;
    v4i g2 = {0, 0, 0, 0};
    v4i g3 = {0, 0, 0, 0};
    v8i g4 = {0, 0, 0, 0, 0, 0, 0, 0};           // unused trailing group
    __builtin_amdgcn_tensor_load_to_lds(g0, g1, g2, g3, g4, 0);
}
__device__ __forceinline__ unsigned lds_offset(const void* p) {
    // generic pointer to LDS: addr[31:0] is the LDS byte address
    return (unsigned)(uintptr_t)p;
}
#endif

// ---------------------------------------------------------------------------
// Grouped 1x1 conv as GEMM. Block: 256 thr = 8 waves, tile 64(co) x 128(l).
// Wave (cw, wq) computes a 32x32 sub-tile as 2x2 WMMA tiles. K staged in
// 64-chunks via double-buffered TDM: chunk ch+1 DMAs while ch computes.
// ---------------------------------------------------------------------------
__global__ __launch_bounds__(256) void conv1x1_wmma_kernel(
    const float* __restrict__ X,     // [N, Cin, L]
    const float* __restrict__ W,     // [Cout, CinG] row-major
    const float* __restrict__ Bias,  // [Cout]
    const float* __restrict__ Res,   // nullptr or [N, Cout, L]
    float* __restrict__ Y,           // [N, Cout, L]
    int Cin, int CinG, int Cout, int CoutG)
{
    const int tid  = threadIdx.x;
    const int wave = tid >> 5;
    const int lane = tid & 31;
    const int half = lane >> 4;
    const int lm   = lane & 15;
    const int cw   = wave >> 2;       // 0..1 : co half (32 rows)
    const int wq   = wave & 3;        // 0..3 : l quarter (32 cols)

    const int lBlk   = blockIdx.x * 128;
    const int coBase = blockIdx.y * 64;
    const int n      = blockIdx.z;
    const int g      = coBase / CoutG;
    const int cinB   = g * CinG;

    __shared__ float Xs[2][64][132];  // double-buffered 64 K x 128 l (+4 pad)
    __shared__ float Ws[2][64][68];   // double-buffered 64 co x 64 K (+4 pad)

    v8f acc00 = {}, acc01 = {}, acc10 = {}, acc11 = {};

    const float* Xbase = X + (size_t)n * Cin * LSEQ + lBlk;
    const float* Wbase = W + (size_t)coBase * CinG;
    const int nChunks = CinG / 64;

#ifdef USE_TDM
    if (wave == 0) {
        // prologue: chunk 0 into buffer 0
        tdm_load_2d_f32(lds_offset(&Xs[0][0][0]),
                        Xbase + (size_t)cinB * LSEQ,
                        128u, 64u, (unsigned)LSEQ, 6u, 3u);
        tdm_load_2d_f32(lds_offset(&Ws[0][0][0]),
                        Wbase, 64u, 64u, (unsigned)CinG, 5u, 3u);
    }
#endif

    for (int ch = 0; ch < nChunks; ++ch) {
        const int buf = ch & 1;
#ifdef USE_TDM
        if (wave == 0) {
            if (ch + 1 < nChunks) {
                const int nb = (ch + 1) & 1;
                const int kB = (ch + 1) * 64;
                tdm_load_2d_f32(lds_offset(&Xs[nb][0][0]),
                                Xbase + (size_t)(cinB + kB) * LSEQ,
                                128u, 64u, (unsigned)LSEQ, 6u, 3u);
                tdm_load_2d_f32(lds_offset(&Ws[nb][0][0]),
                                Wbase + kB, 64u, 64u, (unsigned)CinG, 5u, 3u);
                __builtin_amdgcn_s_wait_tensorcnt(2);  // chunk ch complete
            } else {
                __builtin_amdgcn_s_wait_tensorcnt(0);
            }
        }
#else
        {
            const int kB = ch * 64;
            for (int i = tid; i < 64 * 32; i += 256) {
                int r  = i >> 5;
                int c4 = (i & 31) << 2;
                float4 vv = *(const float4*)(Xbase +
                    (size_t)(cinB + kB + r) * LSEQ + c4);
                Xs[buf][r][c4 + 0] = vv.x; Xs[buf][r][c4 + 1] = vv.y;
                Xs[buf][r][c4 + 2] = vv.z; Xs[buf][r][c4 + 3] = vv.w;
            }
            for (int i = tid; i < 64 * 64; i += 256) {
                int r = i >> 6, c = i & 63;
                Ws[buf][r][c] = Wbase[(size_t)r * CinG + kB + c];
            }
        }
#endif
        __syncthreads();    // chunk ch visible to all waves

        const int ra0 = cw * 32 + lm, ra1 = ra0 + 16;
        const int cb0 = wq * 32 + lm, cb1 = cb0 + 16;
        #pragma unroll
        for (int kk = 0; kk < 64; kk += 4) {
            const int k0 = kk + 2 * half;
            v2f a0, a1, b0, b1;
            a0.x = Ws[buf][ra0][k0]; a0.y = Ws[buf][ra0][k0 + 1];
            a1.x = Ws[buf][ra1][k0]; a1.y = Ws[buf][ra1][k0 + 1];
            b0.x = Xs[buf][k0][cb0]; b0.y = Xs[buf][k0 + 1][cb0];
            b1.x = Xs[buf][k0][cb1]; b1.y = Xs[buf][k0 + 1][cb1];
            acc00 = wmma_f32(a0, b0, acc00);
            acc01 = wmma_f32(a0, b1, acc01);
            acc10 = wmma_f32(a1, b0, acc10);
            acc11 = wmma_f32(a1, b1, acc11);
        }
        __syncthreads();    // all waves done with buf before it is re-filled
    }

    // Epilogue: C/D layout row = i + 8*half, col = lm (per 16x16 tile)
    const int c0 = lBlk + wq * 32 + lm;
    #pragma unroll
    for (int i = 0; i < 8; ++i) {
        const int r0 = coBase + cw * 32 + i + 8 * half;
        const int r1 = r0 + 16;
        size_t i00 = ((size_t)n * Cout + r0) * LSEQ + c0;
        size_t i01 = i00 + 16;
        size_t i10 = ((size_t)n * Cout + r1) * LSEQ + c0;
        size_t i11 = i10 + 16;
        float v00 = acc00[i] + Bias[r0];
        float v01 = acc01[i] + Bias[r0];
        float v10 = acc10[i] + Bias[r1];
        float v11 = acc11[i] + Bias[r1];
        if (Res) {
            v00 += Res[i00]; v01 += Res[i01];
            v10 += Res[i10]; v11 += Res[i11];
        }
        Y[i00] = v00; Y[i01] = v01; Y[i10] = v10; Y[i11] = v11;
    }
}

// ---------------------------------------------------------------------------
// Flash-style attention (softmax over the context axis l):
//   S[l,m] = sum_d k[d,l] q[d,m] / sqrt(L);  P = softmax_l(S);  O = V @ P
// 4 waves/block share 64-wide staged K (32x64) / V (64x64) tiles (row stride
// 81: odd -> row-varying reads conflict-free; 2*81 mod 64 = 34 -> half-split
// reads conflict-free). 4 sub-steps per staging; 2 barriers per 96 WMMAs.
// Writes T = tokens + attn.
// ---------------------------------------------------------------------------
__global__ __launch_bounds__(128) void attention_wmma_kernel(
    const float* __restrict__ Km,   // [N, 512, L]
    const float* __restrict__ Qm,   // [N, 512, L]
    const float* __restrict__ Vm,   // [N, 1024, L]
    const float* __restrict__ Tok,  // [N, 1024, L]
    float* __restrict__ T)          // [N, 1024, L]
{
    const int tid  = threadIdx.x;
    const int wave = tid >> 5;
    const int lane = tid & 31;
    const int half = lane >> 4;
    const int lm   = lane & 15;

    const int m0 = (blockIdx.x * 4 + wave) * 16;
    const int h  = blockIdx.y;
    const int n  = blockIdx.z;

    const float* kp = Km + ((size_t)n * (CTC / 2) + h * DKH) * LSEQ;
    const float* qp = Qm + ((size_t)n * (CTC / 2) + h * DKH) * LSEQ;
    const float* vp = Vm + ((size_t)n * CTC + h * DVH) * LSEQ;

    __shared__ float Ks[DKH][81];       // K tile  [d][l_local 0..63]
    __shared__ float Vs[DVH][81];       // V tile  [c][l_local 0..63]
    __shared__ float Ps[4][16][17];     // per-wave P tile [l_local][m]
    float (* const P)[17] = Ps[wave];

    // Hoist Q B-fragments (loop-invariant): B[K=d][N=m]
    v2f qb[8];
    #pragma unroll
    for (int kk = 0; kk < 8; ++kk) {
        const int d0 = kk * 4 + 2 * half;
        qb[kk].x = qp[(size_t)d0 * LSEQ + m0 + lm];
        qb[kk].y = qp[(size_t)(d0 + 1) * LSEQ + m0 + lm];
    }

    v8f o0 = {}, o1 = {}, o2 = {}, o3 = {};
    float mrun = -3.402823466e38f;
    float srun = 0.0f;
    const float rscale = 0.022097086912079610f;   // 1/sqrt(2048)

    for (int lb = 0; lb < LSEQ; lb += 64) {
        // ---- stage K (32x64) and V (64x64) tiles, block-wide
        for (int i = tid; i < DKH * 16; i += 128) {
            int r = i >> 4, c4 = (i & 15) << 2;
            const float* src = kp + (size_t)r * LSEQ + lb + c4;
            if (lb + 64 < LSEQ) __builtin_prefetch(src + 64, 0, 0);
            float4 vv = *(const float4*)src;
            Ks[r][c4 + 0] = vv.x; Ks[r][c4 + 1] = vv.y;
            Ks[r][c4 + 2] = vv.z; Ks[r][c4 + 3] = vv.w;
        }
        for (int i = tid; i < DVH * 16; i += 128) {
            int r = i >> 4, c4 = (i & 15) << 2;
            const float* src = vp + (size_t)r * LSEQ + lb + c4;
            if (lb + 64 < LSEQ) __builtin_prefetch(src + 64, 0, 0);
            float4 vv = *(const float4*)src;
            Vs[r][c4 + 0] = vv.x; Vs[r][c4 + 1] = vv.y;
            Vs[r][c4 + 2] = vv.z; Vs[r][c4 + 3] = vv.w;
        }
        __syncthreads();

        #pragma unroll
        for (int sub = 0; sub < 64; sub += 16) {
            // ---- S tile = K^T Q : A[M=l][K=d] from Ks, B from registers
            v8f s = {};
            #pragma unroll
            for (int kk = 0; kk < 8; ++kk) {
                const int d0 = kk * 4 + 2 * half;
                v2f a;
                a.x = Ks[d0][sub + lm];
                a.y = Ks[d0 + 1][sub + lm];
                s = wmma_f32(a, qb[kk], s);
            }

            // ---- online softmax over l (per column m; halves share col lm)
            float tmax = -3.402823466e38f;
            #pragma unroll
            for (int i = 0; i < 8; ++i) {
                float v = s[i] * rscale;
                s[i] = v;
                tmax = fmaxf(tmax, v);
            }
            tmax = fmaxf(tmax, __shfl_xor(tmax, 16));
            const float mnew = fmaxf(mrun, tmax);
            const float corr = expf(mrun - mnew);
            mrun = mnew;

            float tsum = 0.0f;
            #pragma unroll
            for (int i = 0; i < 8; ++i) {
                float p = expf(s[i] - mnew);
                s[i] = p;
                tsum += p;
            }
            tsum += __shfl_xor(tsum, 16);
            srun = srun * corr + tsum;

            #pragma unroll
            for (int i = 0; i < 8; ++i) {
                o0[i] *= corr; o1[i] *= corr; o2[i] *= corr; o3[i] *= corr;
            }

            // ---- stage P (wave-private LDS; per-wave DS is in-order, so
            //      no barrier needed between write and read)
            #pragma unroll
            for (int i = 0; i < 8; ++i) P[i + 8 * half][lm] = s[i];

            // ---- O += V @ P : A[M=c][K=l_local] from Vs, B from P
            #pragma unroll
            for (int kk = 0; kk < 16; kk += 4) {
                const int l0 = kk + 2 * half;
                v2f b;
                b.x = P[l0][lm];
                b.y = P[l0 + 1][lm];
                v2f a;
                a.x = Vs[lm][sub + l0];      a.y = Vs[lm][sub + l0 + 1];
                o0 = wmma_f32(a, b, o0);
                a.x = Vs[lm + 16][sub + l0]; a.y = Vs[lm + 16][sub + l0 + 1];
                o1 = wmma_f32(a, b, o1);
                a.x = Vs[lm + 32][sub + l0]; a.y = Vs[lm + 32][sub + l0 + 1];
                o2 = wmma_f32(a, b, o2);
                a.x = Vs[lm + 48][sub + l0]; a.y = Vs[lm + 48][sub + l0 + 1];
                o3 = wmma_f32(a, b, o3);
            }
        }
        __syncthreads();   // all waves done with Ks/Vs before next staging
    }

    // ---- finalize: divide by softmax denominator, add token residual
    const float inv = 1.0f / srun;
    const int col = m0 + lm;
    const size_t base = ((size_t)n * CTC + h * DVH) * LSEQ;
    #pragma unroll
    for (int i = 0; i < 8; ++i) {
        const int r0 = i + 8 * half;
        size_t i0 = base + (size_t)r0 * LSEQ + col;
        T[i0] = Tok[i0] + o0[i] * inv;
        size_t i1 = base + (size_t)(r0 + 16) * LSEQ + col;
        T[i1] = Tok[i1] + o1[i] * inv;
        size_t i2 = base + (size_t)(r0 + 32) * LSEQ + col;
        T[i2] = Tok[i2] + o2[i] * inv;
        size_t i3 = base + (size_t)(r0 + 48) * LSEQ + col;
        T[i3] = Tok[i3] + o3[i] * inv;
    }
}

// ---------------------------------------------------------------------------
extern "C" void kernel_launch(void* const* d_in, const int* in_sizes, int n_in,
                              void* d_out, int out_size, void* d_ws, size_t ws_size,
                              hipStream_t stream) {
    (void)in_sizes; (void)n_in; (void)out_size; (void)ws_size;
    const float* tokens = (const float*)d_in[0];
    const float* k_w    = (const float*)d_in[1];
    const float* k_b    = (const float*)d_in[2];
    const float* q_w    = (const float*)d_in[3];
    const float* q_b    = (const float*)d_in[4];
    const float* v_w    = (const float*)d_in[5];
    const float* v_b    = (const float*)d_in[6];
    const float* ff_w   = (const float*)d_in[7];
    const float* ff_b   = (const float*)d_in[8];
    float* out = (float*)d_out;

    // Workspace layout (48 MB total): k(8MB) q(8MB) v(16MB) t(16MB)
    float* kbuf = (float*)d_ws;
    float* qbuf = kbuf + (size_t)NBAT * (CTC / 2) * LSEQ;
    float* vbuf = qbuf + (size_t)NBAT * (CTC / 2) * LSEQ;
    float* tbuf = vbuf + (size_t)NBAT * CTC * LSEQ;

    const dim3 blk(256);
    conv1x1_wmma_kernel<<<dim3(LSEQ / 128, 512 / 64, NBAT), blk, 0, stream>>>(
        tokens, k_w, k_b, nullptr, kbuf, CTC, 128, 512, 64);
    conv1x1_wmma_kernel<<<dim3(LSEQ / 128, 512 / 64, NBAT), blk, 0, stream>>>(
        tokens, q_w, q_b, nullptr, qbuf, CTC, 128, 512, 64);
    conv1x1_wmma_kernel<<<dim3(LSEQ / 128, 1024 / 64, NBAT), blk, 0, stream>>>(
        tokens, v_w, v_b, nullptr, vbuf, CTC, 128, 1024, 128);
    attention_wmma_kernel<<<dim3(LSEQ / 64, HEADS, NBAT), dim3(128), 0, stream>>>(
        kbuf, qbuf, vbuf, tokens, tbuf);
    conv1x1_wmma_kernel<<<dim3(LSEQ / 128, 1024 / 64, NBAT), blk, 0, stream>>>(
        tbuf, ff_w, ff_b, tbuf, out, CTC, CTC, CTC, CTC);
}